// CPPScatterOpModule_6640019440385
// MI455X (gfx1250) — compile-verified
//
#include <hip/hip_runtime.h>
#include <stdint.h>

#define F_IN    4
#define R       1024
#define C       4096
#define N_CELLS 8192
#define BLOCK   1024   // 32 wave32s = 8 waves/SIMD32: max latency hiding for the
                       // single 192KB-LDS workgroup resident on each WGP

// LDS layout (dynamic shared, 196608 bytes total):
//   [0            , 8*C floats)  : output accumulators, channels 0..7 for this r
//   [8*C          , 12*C floats) : staged input rows in[f][r][0..C) for f=0..3
#define OUT_S_FLOATS (8 * C)
#define IN_S_FLOATS  (F_IN * C)
#define SMEM_BYTES   ((OUT_S_FLOATS + IN_S_FLOATS) * sizeof(float))

typedef unsigned int v4u __attribute__((ext_vector_type(4)));
typedef int          v8i __attribute__((ext_vector_type(8)));
typedef int          v4i __attribute__((ext_vector_type(4)));

__global__ void __launch_bounds__(BLOCK)
scatter3_rowlds_kernel(const float* __restrict__ in,
                       const int* __restrict__ idx0,
                       const int* __restrict__ idx1,
                       const int* __restrict__ idx2,
                       float* __restrict__ out) {
    extern __shared__ float smem[];
    float* out_s = smem;                 // 8*4096 f32 accumulators
    float* in_s  = smem + OUT_S_FLOATS;  // 4*4096 f32 staged input

    const int r   = blockIdx.x;
    const int tid = threadIdx.x;

    // ---- zero the LDS accumulators (b128 stores) ----
    {
        float4  z = make_float4(0.f, 0.f, 0.f, 0.f);
        float4* o4 = (float4*)out_s;
        #pragma unroll 4
        for (int j = tid; j < OUT_S_FLOATS / 4; j += BLOCK) o4[j] = z;
    }

    // ---- TDM: one 2-D tensor tile load, 4 rows x 4096 f32, row stride R*C ----
    // Issued by wave 0 only; TDM ignores EXEC, tracked by TENSORcnt.
#if defined(__has_builtin) && __has_builtin(__builtin_amdgcn_tensor_load_to_lds)
    if (tid < 32) {
        unsigned long long gaddr =
            (unsigned long long)(uintptr_t)(in + (size_t)r * C);
        unsigned lds_off = (unsigned)(uintptr_t)in_s;  // flat low 32 bits = LDS byte offset

        v4u g0;
        g0.x = 1u;                                        // count=1, user descriptor
        g0.y = lds_off;                                   // lds_addr (bytes)
        g0.z = (unsigned)(gaddr & 0xFFFFFFFFull);         // global_addr[31:0]
        g0.w = (unsigned)((gaddr >> 32) & 0x1FFFFFFull)   // global_addr[56:32]
             | (2u << 30);                                // type = 2 ("image")

        v8i g1;
        g1[0] = (int)(2u << 16);               // data_size = 2 (4 bytes); wg_mask=0
        g1[1] = (int)((unsigned)C << 16);      // tensor_dim0[15:0] = 4096
        g1[2] = (int)((unsigned)F_IN << 16);   // tensor_dim0[31:16]=0 | tensor_dim1 = 4
        g1[3] = (int)((unsigned)C << 16);      // tensor_dim1[31:16]=0 | tile_dim0 = 4096
        g1[4] = F_IN;                          // tile_dim1 = 4, tile_dim2 = 0
        g1[5] = R * C;                         // tensor_dim0_stride = 4194304 elems
        g1[6] = 0;                             // stride0[47:32]=0, stride1[15:0]=0
        g1[7] = 0;

        v4i gz4 = {0, 0, 0, 0};
        v8i gz8 = {0, 0, 0, 0, 0, 0, 0, 0};
        // 6-arg variant (clang-23 / therock-10.0 headers):
        //   (v4u group0, v8i group1, v4i group2, v4i group3, v8i extra, i32 cpol)
        __builtin_amdgcn_tensor_load_to_lds(g0, g1, gz4, gz4, gz8, 0);
        __builtin_amdgcn_s_wait_tensorcnt(0);
    }
#else
    // Fallback: cooperative b128 copy of the 4 feature rows into LDS.
    for (int j = tid; j < IN_S_FLOATS / 4; j += BLOCK) {
        int f  = j >> 10;          // 1024 float4 per row
        int c4 = j & 1023;
        float4 v = *(const float4*)(in + ((size_t)f * R + r) * C + (size_t)c4 * 4);
        *(float4*)(in_s + f * C + c4 * 4) = v;
    }
#endif
    __syncthreads();

    // ---- gather from LDS, multiply, scatter-add into LDS accumulators ----
    for (int i = tid; i < N_CELLS; i += BLOCK) {
        int c0 = idx0[i];
        int c1 = idx1[i];
        int c2 = idx2[i];
        #pragma unroll
        for (int f = 0; f < F_IN; ++f) {
            float a = in_s[f * C + c0];
            float b = in_s[f * C + c1];
            float c = in_s[f * C + c2];
            float ab  = a * b;
            float bc  = b * c;
            float ac  = a * c;
            float abc = ab * c;
            // part0: mp3 scattered to all three index sets
            atomicAdd(&out_s[f * C + c0], abc);
            atomicAdd(&out_s[f * C + c1], abc);
            atomicAdd(&out_s[f * C + c2], abc);
            // part1: pairwise products
            atomicAdd(&out_s[(F_IN + f) * C + c0], bc);
            atomicAdd(&out_s[(F_IN + f) * C + c1], ac);
            atomicAdd(&out_s[(F_IN + f) * C + c2], ab);
        }
    }
    __syncthreads();

    // ---- write back 8 channels x 4096 f32 with b128 stores ----
    for (int j = tid; j < OUT_S_FLOATS / 4; j += BLOCK) {
        int ch = j >> 10;          // 1024 float4 per channel row
        int c4 = j & 1023;
        float4 v = *(float4*)(out_s + ch * C + c4 * 4);
        *(float4*)(out + ((size_t)ch * R + r) * C + (size_t)c4 * 4) = v;
    }
}

extern "C" void kernel_launch(void* const* d_in, const int* in_sizes, int n_in,
                              void* d_out, int out_size, void* d_ws, size_t ws_size,
                              hipStream_t stream) {
    (void)in_sizes; (void)n_in; (void)out_size; (void)d_ws; (void)ws_size;

    const float* in  = (const float*)d_in[0];
    const int*   i0  = (const int*)d_in[1];
    const int*   i1  = (const int*)d_in[2];
    const int*   i2  = (const int*)d_in[3];
    float*       out = (float*)d_out;

    // 192 KB dynamic LDS per workgroup (within CDNA5's 320 KB/WGP).
    (void)hipFuncSetAttribute((const void*)scatter3_rowlds_kernel,
                              hipFuncAttributeMaxDynamicSharedMemorySize,
                              (int)SMEM_BYTES);

    scatter3_rowlds_kernel<<<R, BLOCK, SMEM_BYTES, stream>>>(in, i0, i1, i2, out);
}